// FilteredBackProjection_29961691857455
// MI455X (gfx1250) — compile-verified
//
#include <hip/hip_runtime.h>
#include <math.h>

// Problem constants (from reference)
#define D_DET   363     // detectors
#define N_ANG   180     // angles
#define N_BATCH 4
#define IMG     256
#define M_ROWS  (N_BATCH * N_ANG)   // 720 GEMM rows (batch*angles)
#define KPAD    364     // K padded to multiple of 4 (WMMA f32 K-step)
#define NPAD    368     // N padded to multiple of 16
#define ACHUNK  36      // angles per LDS chunk (36*368*4 = 53KB)

typedef float v2f __attribute__((ext_vector_type(2)));
typedef float v8f __attribute__((ext_vector_type(8)));

// ---------------------------------------------------------------------------
// h[n] = real(ifft(2*|fftfreq(D)|))[n] = (4/D^2) * sum_{k=1}^{(D-1)/2} k*cos(2*pi*k*n/D)
// Computed in double for exactness (363*181 cos evals, trivial).
// ---------------------------------------------------------------------------
__global__ void compute_ramp_kernel(float* __restrict__ h) {
    int d = blockIdx.x * blockDim.x + threadIdx.x;
    if (d >= D_DET) return;
    const double w = 2.0 * M_PI * (double)d / (double)D_DET;
    double acc = 0.0;
    for (int k = 1; k <= (D_DET - 1) / 2; ++k)
        acc += (double)k * cos(w * (double)k);
    h[d] = (float)(acc * (4.0 / ((double)D_DET * (double)D_DET)));
}

// Zero-padded copy of sinogram: Apad[r][k], 720 x 364 (col 363 = 0)
__global__ void build_apad_kernel(const float* __restrict__ sino, float* __restrict__ Apad) {
    int idx = blockIdx.x * blockDim.x + threadIdx.x;
    if (idx >= M_ROWS * KPAD) return;
    int r = idx / KPAD, k = idx - r * KPAD;
    Apad[idx] = (k < D_DET) ? sino[r * D_DET + k] : 0.0f;
}

// Circulant filter matrix: Bpad[k][d] = h[(d-k) mod 363], 364 x 368, zero padded
__global__ void build_bpad_kernel(const float* __restrict__ h, float* __restrict__ Bpad) {
    int idx = blockIdx.x * blockDim.x + threadIdx.x;
    if (idx >= KPAD * NPAD) return;
    int k = idx / NPAD, d = idx - k * NPAD;
    float v = 0.0f;
    if (k < D_DET && d < D_DET) {
        int m = d - k;
        if (m < 0) m += D_DET;
        v = h[m];
    }
    Bpad[idx] = v;
}

// ---------------------------------------------------------------------------
// Ramp filtering as fp32 WMMA GEMM: Cpad(720x368) = Apad(720x364) x Bpad(364x368)
// One wave per 16x16 output tile. V_WMMA_F32_16X16X4_F32, K loop of 91 steps.
// A frag (16x4 f32): lane<16 -> M=lane, K=k0..k0+1 ; lane>=16 -> M=lane-16, K=k0+2..k0+3
// B frag (4x16 f32): mirrored with N in place of M.
// C/D (16x16 f32): vgpr j -> M = (lane<16 ? j : j+8), N = lane&15.
// ---------------------------------------------------------------------------
__global__ void __launch_bounds__(32)
ramp_gemm_wmma_kernel(const float* __restrict__ Apad,
                      const float* __restrict__ Bpad,
                      float* __restrict__ Cpad) {
    const int lane = threadIdx.x & 31;
    const int half = lane >> 4;      // 0: K pair {0,1}, 1: K pair {2,3}
    const int l15  = lane & 15;
    const int ntile = blockIdx.x;    // 0..22
    const int mtile = blockIdx.y;    // 0..44

    const int row = mtile * 16 + l15;
    const int col = ntile * 16 + l15;
    const float* __restrict__ Arow = Apad + (size_t)row * KPAD + half * 2;
    const float* __restrict__ Bcol = Bpad + col + (size_t)(half * 2) * NPAD;

    v8f acc = {0.f, 0.f, 0.f, 0.f, 0.f, 0.f, 0.f, 0.f};

    for (int k0 = 0; k0 < KPAD; k0 += 4) {
        v2f a;
        a.x = Arow[k0];
        a.y = Arow[k0 + 1];
        v2f b;
        b.x = Bcol[(size_t)k0 * NPAD];
        b.y = Bcol[(size_t)(k0 + 1) * NPAD];
        acc = __builtin_amdgcn_wmma_f32_16x16x4_f32(
            /*neg_a=*/false, a, /*neg_b=*/false, b,
            /*c_mod=*/(short)0, acc, /*reuse_a=*/false, /*reuse_b=*/false);
    }

    float* __restrict__ Crow = Cpad + (size_t)(mtile * 16 + half * 8) * NPAD + col;
#pragma unroll
    for (int j = 0; j < 8; ++j)
        Crow[(size_t)j * NPAD] = acc[j];
}

// ---------------------------------------------------------------------------
// Backprojection: one block = one image row (iy) of one batch; thread = pixel x.
// Filtered sinogram streamed through LDS in 36-angle chunks (53KB static LDS).
// ---------------------------------------------------------------------------
__global__ void __launch_bounds__(256)
backproject_kernel(const float* __restrict__ Cpad, float* __restrict__ out) {
    __shared__ float s_sino[ACHUNK * NPAD];
    __shared__ float s_cos[N_ANG];
    __shared__ float s_sin[N_ANG];

    const int tid = threadIdx.x;          // pixel x (0..255)
    const int iy  = blockIdx.x;           // image row
    const int b   = blockIdx.y;           // batch

    if (tid < N_ANG) {
        float ang = 3.14159265358979323846f * (float)tid / (float)(N_ANG - 1);
        s_cos[tid] = cosf(ang);
        s_sin[tid] = sinf(ang);
    }

    const float x = -1.0f + 2.0f * (float)tid / (float)(IMG - 1);
    const float y = -1.0f + 2.0f * (float)iy  / (float)(IMG - 1);

    float acc = 0.0f;

    for (int a0 = 0; a0 < N_ANG; a0 += ACHUNK) {
        __syncthreads();   // previous chunk fully consumed (also orders table init)
        const float4* __restrict__ src =
            (const float4*)(Cpad + (size_t)(b * N_ANG + a0) * NPAD);
        float4* __restrict__ dst = (float4*)s_sino;
        for (int i = tid; i < (ACHUNK * NPAD) / 4; i += 256)
            dst[i] = src[i];
        __syncthreads();

#pragma unroll 4
        for (int a = 0; a < ACHUNK; ++a) {
            const float c = s_cos[a0 + a];
            const float s = s_sin[a0 + a];
            const float t  = x * c + y * s;
            const float px = (t + 1.0f) * 0.5f * (float)(D_DET - 1);
            const float fi = floorf(px);
            const int   i0 = (int)fi;
            const float frac = px - fi;
            const int   i1 = i0 + 1;
            const float w0 = (i0 >= 0 && i0 <= D_DET - 1) ? (1.0f - frac) : 0.0f;
            const float w1 = (i1 >= 0 && i1 <= D_DET - 1) ? frac : 0.0f;
            const int i0c = min(max(i0, 0), D_DET - 1);
            const int i1c = min(max(i1, 0), D_DET - 1);
            const float* __restrict__ rowp = s_sino + a * NPAD;
            acc = fmaf(rowp[i0c], w0, acc);
            acc = fmaf(rowp[i1c], w1, acc);
        }
    }

    out[((size_t)b * IMG + iy) * IMG + tid] =
        acc * (3.14159265358979323846f / (float)N_ANG);
}

// ---------------------------------------------------------------------------
extern "C" void kernel_launch(void* const* d_in, const int* in_sizes, int n_in,
                              void* d_out, int out_size, void* d_ws, size_t ws_size,
                              hipStream_t stream) {
    (void)in_sizes; (void)n_in; (void)out_size; (void)ws_size;

    const float* sino = (const float*)d_in[0];   // (4,1,180,363) fp32
    float* out = (float*)d_out;                  // (4,1,256,256) fp32

    // Workspace layout (floats): h | Apad | Bpad | Cpad  (~2.7 MB total)
    float* ws   = (float*)d_ws;
    float* h    = ws;                                     // 363 (+pad)
    float* Apad = ws + 512;                               // 720*364
    float* Bpad = Apad + (size_t)M_ROWS * KPAD;           // 364*368
    float* Cpad = Bpad + (size_t)KPAD * NPAD;             // 720*368

    compute_ramp_kernel<<<dim3((D_DET + 127) / 128), 128, 0, stream>>>(h);
    build_apad_kernel<<<dim3((M_ROWS * KPAD + 255) / 256), 256, 0, stream>>>(sino, Apad);
    build_bpad_kernel<<<dim3((KPAD * NPAD + 255) / 256), 256, 0, stream>>>(h, Bpad);
    ramp_gemm_wmma_kernel<<<dim3(NPAD / 16, M_ROWS / 16), 32, 0, stream>>>(Apad, Bpad, Cpad);
    backproject_kernel<<<dim3(IMG, N_BATCH), 256, 0, stream>>>(Cpad, out);
}